// SyntheticTripletLoss_22832046145612
// MI455X (gfx1250) — compile-verified
//
#include <hip/hip_runtime.h>
#include <hip/hip_bf16.h>

typedef float v2f __attribute__((ext_vector_type(2)));
typedef float v4f __attribute__((ext_vector_type(4)));
typedef float v8f __attribute__((ext_vector_type(8)));

#define TL_D      128   // feature dim (reference fixes D=128)
#define TL_MARGIN 0.5f
#define TL_EPS    1e-12f

// ---------------------------------------------------------------------------
// Kernel 1: streaming triplet loss. One wave32 per row; lane i holds columns
// [4i, 4i+3] of preds/targets as float4. Non-temporal b128 loads (1 GiB
// streamed once >> 192 MB L2, so keep it out of L2). Butterfly reductions
// across the wave (warpSize==32 on gfx1250). Per-block partial sums -> d_ws.
// Roofline: ~1.07 GiB / 23.3 TB/s ~= 46 us; VALU need (~17.5 TFLOP/s at full
// BW) is far below peak, so this stays bandwidth-bound.
// ---------------------------------------------------------------------------
__global__ __launch_bounds__(256) void triplet_loss_main(
    const float* __restrict__ targets, const float* __restrict__ preds,
    float* __restrict__ partials, int nrows)
{
    const int lane          = threadIdx.x & 31;
    const int wave          = threadIdx.x >> 5;          // 0..7
    const int wavesPerBlock = blockDim.x >> 5;
    const int gwave         = blockIdx.x * wavesPerBlock + wave;
    const int nwaves        = gridDim.x * wavesPerBlock;

    float acc = 0.0f;

    for (int row = gwave; row < nrows; row += nwaves) {
        const v4f* tp = (const v4f*)(targets + (size_t)row * TL_D) + lane;
        const v4f* pp = (const v4f*)(preds   + (size_t)row * TL_D) + lane;
        v4f tv = __builtin_nontemporal_load(tp);   // global_load_b128 th:NT
        v4f pv = __builtin_nontemporal_load(pp);

        // d = p . t  (wave butterfly -> all lanes)
        float d = pv.x * tv.x + pv.y * tv.y + pv.z * tv.z + pv.w * tv.w;
        #pragma unroll
        for (int m = 16; m >= 1; m >>= 1) d += __shfl_xor(d, m, 32);

        // neg = p - d*t, formed in registers (same order as reference)
        v4f ng = pv - tv * d;

        float nn = ng.x * ng.x + ng.y * ng.y + ng.z * ng.z + ng.w * ng.w;
        float sp = pv.x * ng.x + pv.y * ng.y + pv.z * ng.z + pv.w * ng.w;
        #pragma unroll
        for (int m = 16; m >= 1; m >>= 1) {
            nn += __shfl_xor(nn, m, 32);
            sp += __shfl_xor(sp, m, 32);
        }

        float norm = fmaxf(sqrtf(fmaxf(nn, 0.0f)), TL_EPS);
        float sim  = sp / norm;
        float loss = fmaxf(TL_MARGIN + sim - d, 0.0f);
        acc += loss;   // identical on all lanes; only lane 0's copy is used
    }

    __shared__ float ssum[8];
    if (lane == 0) ssum[wave] = acc;
    __syncthreads();
    if (threadIdx.x == 0) {
        float s = 0.0f;
        for (int i = 0; i < wavesPerBlock; ++i) s += ssum[i];
        partials[blockIdx.x] = s;
    }
}

// ---------------------------------------------------------------------------
// Kernel 2: single-wave final reduction via v_wmma_f32_16x16x4_f32.
// B = all-ones (4x16) so D[m][n] = C[m][n] + sum_k A[m][k]; the accumulator C
// carries the running sum across 64-element chunks in full fp32. A-matrix f32
// 16x4 layout: lane l<16 holds (K=0,K=1) of row M=l, lane l>=16 holds (K=2,K=3)
// of row M=l-16 -> each lane loads one b64 at 4*(lane%16)+2*(lane/16).
// Full chunks are loaded unconditionally (no per-lane EXEC churn around the
// WMMA); only a uniform tail branch (n%64 != 0, never hit in the reference
// config) uses per-lane guarded loads. EXEC is all-ones at every WMMA: the
// block is exactly one wave and the tail condition is wave-uniform.
// C/D layout: lane l, VGPR v holds M = v + 8*(l>=16), N = l%16; the grand
// total is sum_v C[v] on lanes 0 and 16 (column N=0).
// ---------------------------------------------------------------------------
__global__ __launch_bounds__(32) void triplet_loss_reduce_wmma(
    const float* __restrict__ partials, int n, float inv_n,
    float* __restrict__ out)
{
    const int lane = threadIdx.x;        // 0..31
    const int m    = lane & 15;          // A-matrix row
    const int half = lane >> 4;          // K pair: 0 -> (0,1), 1 -> (2,3)
    const int loff = 4 * m + 2 * half;   // lane's element offset within a chunk

    v8f c = {0.f, 0.f, 0.f, 0.f, 0.f, 0.f, 0.f, 0.f};
    const v2f ones = {1.0f, 1.0f};       // every B element = 1.0 (layout-proof)

    const int nfull = n & ~63;           // whole 64-element chunks

    for (int base = 0; base < nfull; base += 64) {
        // 8-byte aligned b64 load, no bounds check needed in full chunks
        v2f a = *(const v2f*)(partials + base + loff);
        // 8 args: (neg_a, A, neg_b, B, c_mod, C, reuse_a, reuse_b)
        c = __builtin_amdgcn_wmma_f32_16x16x4_f32(
                false, a, false, ones, (short)0, c, false, false);
    }

    if (nfull < n) {                     // uniform branch: whole wave enters
        int idx = nfull + loff;
        v2f a;
        a.x = (idx     < n) ? partials[idx]     : 0.0f;
        a.y = (idx + 1 < n) ? partials[idx + 1] : 0.0f;
        c = __builtin_amdgcn_wmma_f32_16x16x4_f32(
                false, a, false, ones, (short)0, c, false, false);
    }

    float s = c[0] + c[1] + c[2] + c[3] + c[4] + c[5] + c[6] + c[7];
    float s_hi = __shfl(s, 16, 32);      // rows 8..15 of column 0
    if (lane == 0) out[0] = (s + s_hi) * inv_n;
}

// ---------------------------------------------------------------------------
extern "C" void kernel_launch(void* const* d_in, const int* in_sizes, int n_in,
                              void* d_out, int out_size, void* d_ws, size_t ws_size,
                              hipStream_t stream)
{
    const float* targets  = (const float*)d_in[0];
    const float* preds    = (const float*)d_in[1];
    float*       out      = (float*)d_out;
    float*       partials = (float*)d_ws;

    const int nrows = in_sizes[0] / TL_D;   // 1048576 for the reference setup

    int nblocks = 2048;                     // 16384 waves; 8 KB of partials
    if ((size_t)nblocks * sizeof(float) > ws_size)
        nblocks = (int)(ws_size / sizeof(float));
    if (nblocks >= 64) nblocks &= ~63;      // keep the reduce loop branchless
    if (nblocks < 1) nblocks = 1;

    triplet_loss_main<<<nblocks, 256, 0, stream>>>(targets, preds, partials, nrows);
    triplet_loss_reduce_wmma<<<1, 32, 0, stream>>>(partials, nblocks,
                                                   1.0f / (float)nrows, out);
}